// SparseConvolutionBase_69097433858537
// MI455X (gfx1250) — compile-verified
//
#include <hip/hip_runtime.h>

#define NVOX  262144
#define MPAIR 131072
#define KOFF  27
#define CIN   32
#define COUT  32

typedef __attribute__((ext_vector_type(2))) float v2f;
typedef __attribute__((ext_vector_type(8))) float v8f;

__global__ void spconv_bias_init(const float* __restrict__ bias,
                                 float* __restrict__ out, int total) {
  int i = blockIdx.x * blockDim.x + threadIdx.x;
  if (i < total) out[i] = bias[i & (COUT - 1)];
}

__global__ __launch_bounds__(256) void spconv_wmma_f32(
    const float* __restrict__ input,   // [N, 32]
    const float* __restrict__ kernel,  // [27, 32, 32]
    const int*   __restrict__ in_map,  // [27, M]
    const int*   __restrict__ out_map, // [27, M]
    float*       __restrict__ out)     // [N, 32]
{
  const int k    = blockIdx.y;            // kernel offset, uniform per block
  const int lane = threadIdx.x & 31;
  const int wave = threadIdx.x >> 5;
  const int half = lane >> 4;             // 0: lanes 0-15, 1: lanes 16-31
  const int l16  = lane & 15;

  const float* Wk = kernel + k * CIN * COUT;

  // B fragments (4x16 fp32 tiles of the weight matrix), loaded once per wave
  // and held in registers across the whole persistent M loop.
  // VGPR0 holds K = 2*half, VGPR1 holds K = 2*half+1; N = l16 (+16 per ntile).
  v2f bfrag[2][8];
  #pragma unroll
  for (int n = 0; n < 2; ++n) {
    #pragma unroll
    for (int kk = 0; kk < 8; ++kk) {
      const int krow = kk * 4 + 2 * half;
      v2f b;
      b.x = Wk[(krow    ) * COUT + n * 16 + l16];
      b.y = Wk[(krow + 1) * COUT + n * 16 + l16];
      bfrag[n][kk] = b;
    }
  }

  const int tilesPerK  = MPAIR / 16;                    // 8192
  const int wavesTotal = gridDim.x * (blockDim.x >> 5); // waves per k-offset
  const int gw         = blockIdx.x * (blockDim.x >> 5) + wave;

  for (int t = gw; t < tilesPerK; t += wavesTotal) {    // wave-uniform bound
    const int m0      = t * 16;
    const int row_in  = in_map [k * MPAIR + m0 + l16];  // lanes 16-31 mirror 0-15
    const int row_out = out_map[k * MPAIR + m0 + l16];
    const float* arow = input + (long)row_in * CIN + 2 * half;

    v8f c0 = {};  // c_out 0..15
    v8f c1 = {};  // c_out 16..31
    #pragma unroll
    for (int kk = 0; kk < 8; ++kk) {
      // A fragment: lane<16 holds K = kk*4+{0,1}; lane>=16 holds K = kk*4+{2,3}
      v2f a = *(const v2f*)(arow + kk * 4);
      c0 = __builtin_amdgcn_wmma_f32_16x16x4_f32(false, a, false, bfrag[0][kk],
                                                 (short)0, c0, false, false);
      c1 = __builtin_amdgcn_wmma_f32_16x16x4_f32(false, a, false, bfrag[1][kk],
                                                 (short)0, c1, false, false);
    }

    // Scatter-add: D VGPR v holds row M = v + 8*half, col N = l16 (+16).
    #pragma unroll
    for (int v = 0; v < 8; ++v) {
      const int src = v + 8 * half;            // rows live in lanes 0..15
      const int r   = __shfl(row_out, src, 32);
      float* po = out + (long)r * COUT + l16;
      __hip_atomic_fetch_add(po + 0,  c0[v], __ATOMIC_RELAXED, __HIP_MEMORY_SCOPE_AGENT);
      __hip_atomic_fetch_add(po + 16, c1[v], __ATOMIC_RELAXED, __HIP_MEMORY_SCOPE_AGENT);
    }
  }
}

extern "C" void kernel_launch(void* const* d_in, const int* in_sizes, int n_in,
                              void* d_out, int out_size, void* d_ws, size_t ws_size,
                              hipStream_t stream) {
  const float* input  = (const float*)d_in[0];
  const float* kernel = (const float*)d_in[1];
  const float* bias   = (const float*)d_in[2];
  const int*   in_map = (const int*)d_in[3];
  const int*   out_map= (const int*)d_in[4];
  float* out = (float*)d_out;

  // 1) out = bias (broadcast over rows); d_out is poisoned by the harness.
  {
    const int total = NVOX * COUT;
    const int threads = 256;
    const int blocks = (total + threads - 1) / threads;
    spconv_bias_init<<<blocks, threads, 0, stream>>>(bias, out, total);
  }

  // 2) gather -> WMMA fp32 GEMM -> atomic scatter-add.
  //    grid.y = kernel offset (27); grid.x * 8 waves per offset,
  //    each wave persists over 16 M-tiles reusing its weight fragments.
  {
    dim3 grid(64, KOFF);
    dim3 block(256);
    spconv_wmma_f32<<<grid, block, 0, stream>>>(input, kernel, in_map, out_map, out);
  }
}